// FullAttentionDecode_37417755083577
// MI455X (gfx1250) — compile-verified
//
#include <hip/hip_runtime.h>
#include <math.h>

// ---------------------------------------------------------------------------
// FullAttentionDecode for MI455X (gfx1250, wave32, WMMA + TDM).
// Bandwidth-bound (268 MB fp32 KV stream, ~11.5us @ 23.3 TB/s). Strategy:
// flash-decoding split-K, one wave per 256-position chunk. K/V tiles are
// fetched by the Tensor Data Mover (tensor_load_to_lds, double-buffered,
// TENSORcnt-tracked) with native LDS pitch padding; score and P*V matmuls
// run on v_wmma_f32_16x16x4_f32.
// ---------------------------------------------------------------------------

#define HDIM      1536
#define NUM_HEADS 12
#define NUM_KV    2
#define GQA       6
#define HEAD_DIM  128
#define MAX_CACHE 131072
#define QK_SCALE  0.08838834764831845f  // 128^-0.5

#define CHUNK       256
#define NCHUNK      (MAX_CACHE / CHUNK)   // 512 chunks per kv group
#define TILES       (CHUNK / 16)          // 16 tiles of 16 positions
#define PART_STRIDE 2080                  // 16 m + 16 s + 128*16 O^T floats

// workspace layout (float offsets)
#define WS_QPAD 0        // 2 groups * 16 padded heads * 128 = 4096
#define WS_KNEW 4096     // 2 * 128
#define WS_VNEW 4352     // 2 * 128
#define WS_ATTN 4608     // 12 * 128
#define WS_PART 6144     // 2 * 512 * 2080  (~8.5 MB)

typedef __attribute__((ext_vector_type(2))) float v2f;
typedef __attribute__((ext_vector_type(4))) float v4f;
typedef __attribute__((ext_vector_type(8))) float v8f;
typedef __attribute__((ext_vector_type(4))) unsigned int v4u;
typedef __attribute__((ext_vector_type(4))) int v4i;
typedef __attribute__((ext_vector_type(8))) int v8i;

// ---------------------------------------------------------------------------
// TDM: load one 16x128 fp32 tile (row stride 128 elems in memory) into LDS
// at byte offset lds_off with a padded pitch of 132 floats (pad 4 DWORDs
// every 128 DWORDs). Descriptor per CDNA5 ISA section 8.3/8.4.
// ---------------------------------------------------------------------------
__device__ __forceinline__ void tdm_load_tile16x128(unsigned int lds_off,
                                                    const float* gsrc) {
  unsigned long long ga = (unsigned long long)(uintptr_t)gsrc;
  v4u g0;
  g0.x = 1u;                                  // count=1, user mode, no gather
  g0.y = lds_off;                             // lds_addr (bytes)
  g0.z = (unsigned int)ga;                    // global_addr[31:0]
  g0.w = (unsigned int)(ga >> 32) | (2u << 30); // global_addr[56:32] | type=2

  v8i g1;
  g1[0] = (int)((2u << 16)      // data_size = 4 bytes
              | (1u << 20)      // pad_enable
              | (6u << 22)      // pad_interval: 128 DWORDs
              | (3u << 25));    // pad_amount: 4 DWORDs  -> pitch 132 floats
  g1[1] = (int)(128u << 16);    // tensor_dim0 = 128 (atomic_barrier_addr = 0)
  g1[2] = (int)(16u << 16);     // tensor_dim1 = 16
  g1[3] = (int)(128u << 16);    // tile_dim0 = 128
  g1[4] = (int)16u;             // tile_dim1 = 16, tile_dim2 = 0
  g1[5] = (int)128u;            // tensor_dim0_stride = 128
  g1[6] = 0;
  g1[7] = 0;

  v4i gz = (v4i)0;
#if defined(__clang_major__) && (__clang_major__ >= 23)
  v8i gz8 = (v8i)0;
  __builtin_amdgcn_tensor_load_to_lds(g0, g1, gz, gz, gz8, 0);
#else
  __builtin_amdgcn_tensor_load_to_lds(g0, g1, gz, gz, 0);
#endif
}

// ---------------------------------------------------------------------------
// Kernel A: q/k/v projections (GEMV, 2048 rows of 1536). q is pre-scaled and
// scattered into a zero-padded 16-row-per-group layout for the WMMA B operand.
// ---------------------------------------------------------------------------
__global__ void qkv_kernel(const float* __restrict__ x,
                           const float* __restrict__ Wq,
                           const float* __restrict__ Wk,
                           const float* __restrict__ Wv,
                           float* __restrict__ ws) {
  __shared__ float red[256];
  const int r = blockIdx.x;
  const float* wrow;
  if (r < 1536)      wrow = Wq + (size_t)r * HDIM;
  else if (r < 1792) wrow = Wk + (size_t)(r - 1536) * HDIM;
  else               wrow = Wv + (size_t)(r - 1792) * HDIM;

  float acc = 0.f;
  for (int j = threadIdx.x; j < HDIM; j += 256) acc += wrow[j] * x[j];
  red[threadIdx.x] = acc;
  __syncthreads();
  for (int s = 128; s > 0; s >>= 1) {
    if (threadIdx.x < s) red[threadIdx.x] += red[threadIdx.x + s];
    __syncthreads();
  }
  if (threadIdx.x == 0) {
    float v = red[0];
    if (r < 1536) {
      int h = r >> 7, d = r & 127;
      int g = h / GQA, h6 = h % GQA;
      ws[WS_QPAD + (g * 16 + h6) * 128 + d] = v * QK_SCALE;
    } else if (r < 1792) {
      ws[WS_KNEW + (r - 1536)] = v;
    } else {
      ws[WS_VNEW + (r - 1792)] = v;
    }
  }
}

// ---------------------------------------------------------------------------
// Kernel B: flash-decoding partial attention. One wave32 per (group, chunk).
//   S^T(16seq x 16head) = K_tile(16x128) x Q^T(128x16)   [32 wmma 16x16x4]
//   O^T(128dim x 16head) += V^T(128x16seq) x P^T(16x16)  [32 wmma 16x16x4]
// K/V tiles arrive via TDM (double-buffered, s_wait_tensorcnt). Online
// softmax per head lives in lanes (head == lane%16).
// ---------------------------------------------------------------------------
__global__ __launch_bounds__(32) void decode_kernel(const float* __restrict__ kc,
                                                    const float* __restrict__ vc,
                                                    const int*   __restrict__ posp,
                                                    float*       __restrict__ ws) {
  const int pos   = posp[0];
  const int g     = blockIdx.y;
  const int chunk = blockIdx.x;
  const int lane  = threadIdx.x;
  const int col   = lane & 15;   // head (C-layout column)
  const int hi    = lane >> 4;   // half-wave select

  __shared__ __align__(16) float Qs[16 * 132];
  __shared__ __align__(16) float Ks[2][16 * 132];
  __shared__ __align__(16) float Vs[2][16 * 132];
  __shared__ __align__(16) float Ps[16 * 20];

  const float* kg   = kc + (size_t)g * MAX_CACHE * HEAD_DIM;
  const float* vg   = vc + (size_t)g * MAX_CACHE * HEAD_DIM;
  const float* knew = ws + WS_KNEW + g * HEAD_DIM;
  const float* vnew = ws + WS_VNEW + g * HEAD_DIM;

  // Stage Q^T source (16 padded heads x 128 dims), pitch 132 to dodge banks.
  {
    const float* qp = ws + WS_QPAD + g * 16 * 128;
#pragma unroll
    for (int i = 0; i < 16; ++i) {
      int idx4 = i * 32 + lane;          // 512 float4 tiles
      int row  = idx4 >> 3;
      int c4   = (idx4 & 7) * 4;
      *(v4f*)(&Qs[row * 132 + c4]) = *(const v4f*)(qp + row * 128 + c4);
    }
  }
  __syncthreads();

  // Prologue: TDM prefetch of tile 0 into buffer 0.
  {
    const int tb0 = chunk * CHUNK;
    tdm_load_tile16x128((unsigned int)(uintptr_t)&Ks[0][0],
                        kg + (size_t)tb0 * HEAD_DIM);
    tdm_load_tile16x128((unsigned int)(uintptr_t)&Vs[0][0],
                        vg + (size_t)tb0 * HEAD_DIM);
  }

  float mrun = -3.0e38f;
  float srun = 0.f;
  v8f O[8];
#pragma unroll
  for (int t = 0; t < 8; ++t) O[t] = (v8f)(0.0f);

#pragma unroll 1
  for (int tile = 0; tile < TILES; ++tile) {
    const int cur = tile & 1;
    const int tb  = chunk * CHUNK + tile * 16;

    // Kick off TDM for the next tile, then wait for the current one.
    if (tile + 1 < TILES) {
      const int nb = tb + 16;
      tdm_load_tile16x128((unsigned int)(uintptr_t)&Ks[cur ^ 1][0],
                          kg + (size_t)nb * HEAD_DIM);
      tdm_load_tile16x128((unsigned int)(uintptr_t)&Vs[cur ^ 1][0],
                          vg + (size_t)nb * HEAD_DIM);
      __builtin_amdgcn_s_wait_tensorcnt(2);
    } else {
      __builtin_amdgcn_s_wait_tensorcnt(0);
    }

    // Patch the freshly-projected k/v into the row holding `pos` (uniform).
    if (pos >= tb && pos < tb + 16) {
      const int row = pos - tb;
      const int c   = lane * 4;
      *(v4f*)(&Ks[cur][row * 132 + c]) = *(const v4f*)(knew + c);
      *(v4f*)(&Vs[cur][row * 132 + c]) = *(const v4f*)(vnew + c);
    }
    __syncthreads();

    // ---- S^T = K_tile x Q^T via 32 chained WMMA (K dim 128, 4 per op) ----
    v8f S = (v8f)(0.0f);
    {
      const float* ka = &Ks[cur][col * 132 + hi * 2];  // A-frag: M=col, K pair
      const float* qb = &Qs[col * 132 + hi * 2];       // B-frag: N=col, K pair
#pragma unroll
      for (int kk = 0; kk < 32; ++kk) {
        v2f a = *(const v2f*)(ka + kk * 4);
        v2f b = *(const v2f*)(qb + kk * 4);
        S = __builtin_amdgcn_wmma_f32_16x16x4_f32(false, a, false, b,
                                                  (short)0, S, false, false);
      }
    }

    // ---- online softmax (per head == per lane column) ----
    float sv[8], pv[8];
    float tmax = -3.0e38f;
#pragma unroll
    for (int r = 0; r < 8; ++r) {
      int grow = tb + hi * 8 + r;
      float xs = (grow <= pos) ? S[r] : -3.0e38f;
      sv[r] = xs;
      tmax  = fmaxf(tmax, xs);
    }
    tmax = fmaxf(tmax, __shfl_xor(tmax, 16, 32));
    float mnew = fmaxf(mrun, tmax);
    float resc = __expf(mrun - mnew);
    float tsum = 0.f;
#pragma unroll
    for (int r = 0; r < 8; ++r) {
      float p = (sv[r] > -1.0e38f) ? __expf(sv[r] - mnew) : 0.f;
      pv[r] = p;
      tsum += p;
    }
    tsum += __shfl_xor(tsum, 16, 32);
    srun = srun * resc + tsum;
    mrun = mnew;
#pragma unroll
    for (int t = 0; t < 8; ++t) O[t] *= resc;

    // ---- publish P^T tile to LDS (pitch 20 keeps both half-waves apart) ----
#pragma unroll
    for (int r = 0; r < 8; ++r) Ps[(hi * 8 + r) * 20 + col] = pv[r];
    __syncthreads();

    // ---- O^T += V^T x P^T : 8 dim-tiles x 4 seq-chunks of WMMA ----
#pragma unroll
    for (int dt = 0; dt < 8; ++dt) {
      const float* vb = &Vs[cur][hi * 2 * 132 + dt * 16 + col]; // A: M=dim col
      const float* pb = &Ps[hi * 2 * 20 + col];                 // B: N=head col
      v8f o = O[dt];
#pragma unroll
      for (int kkv = 0; kkv < 4; ++kkv) {
        v2f a, b;
        a.x = vb[(kkv * 4 + 0) * 132];
        a.y = vb[(kkv * 4 + 1) * 132];
        b.x = pb[(kkv * 4 + 0) * 20];
        b.y = pb[(kkv * 4 + 1) * 20];
        o = __builtin_amdgcn_wmma_f32_16x16x4_f32(false, a, false, b,
                                                  (short)0, o, false, false);
      }
      O[dt] = o;
    }
    __syncthreads();
  }

  // ---- write partial (m, s, O^T) for this chunk ----
  float* pp = ws + WS_PART + (size_t)(g * NCHUNK + chunk) * PART_STRIDE;
  if (hi == 0) {
    pp[col]      = mrun;
    pp[16 + col] = srun;
  }
#pragma unroll
  for (int dt = 0; dt < 8; ++dt)
#pragma unroll
    for (int r = 0; r < 8; ++r) {
      int dim = dt * 16 + hi * 8 + r;
      pp[32 + dim * 16 + col] = O[dt][r];
    }
}

// ---------------------------------------------------------------------------
// Kernel C1: combine split-K partials -> attn_out[12][128]
// ---------------------------------------------------------------------------
__global__ void reduce_kernel(float* __restrict__ ws) {
  __shared__ float red[256];
  const int g = blockIdx.x / GQA, h = blockIdx.x % GQA;
  const int tid = threadIdx.x;
  const float* pb = ws + WS_PART + (size_t)g * NCHUNK * PART_STRIDE;

  float lm = -3.0e38f;
  for (int c = tid; c < NCHUNK; c += 256)
    lm = fmaxf(lm, pb[(size_t)c * PART_STRIDE + h]);
  red[tid] = lm;
  __syncthreads();
  for (int s = 128; s > 0; s >>= 1) {
    if (tid < s) red[tid] = fmaxf(red[tid], red[tid + s]);
    __syncthreads();
  }
  const float M = red[0];
  __syncthreads();

  float ls = 0.f;
  for (int c = tid; c < NCHUNK; c += 256) {
    const float* p = pb + (size_t)c * PART_STRIDE;
    ls += p[16 + h] * __expf(p[h] - M);
  }
  red[tid] = ls;
  __syncthreads();
  for (int s = 128; s > 0; s >>= 1) {
    if (tid < s) red[tid] += red[tid + s];
    __syncthreads();
  }
  const float Stot = red[0];
  __syncthreads();

  const int half = tid >> 7, d = tid & 127;
  float acc = 0.f;
  for (int c = half; c < NCHUNK; c += 2) {
    const float* p = pb + (size_t)c * PART_STRIDE;
    acc += p[32 + d * 16 + h] * __expf(p[h] - M);
  }
  red[tid] = acc;
  __syncthreads();
  if (tid < 128)
    ws[WS_ATTN + (g * GQA + h) * 128 + d] = (red[tid] + red[tid + 128]) / Stot;
}

// ---------------------------------------------------------------------------
// Kernel C2: y = Wo @ attn_out (1536x1536 GEMV)
// ---------------------------------------------------------------------------
__global__ void out_proj_kernel(const float* __restrict__ Wo,
                                const float* __restrict__ ws,
                                float* __restrict__ out) {
  __shared__ float red[256];
  const int row = blockIdx.x;
  const float* at = ws + WS_ATTN;
  const float* wr = Wo + (size_t)row * (NUM_HEADS * HEAD_DIM);
  float acc = 0.f;
  for (int j = threadIdx.x; j < NUM_HEADS * HEAD_DIM; j += 256) acc += wr[j] * at[j];
  red[threadIdx.x] = acc;
  __syncthreads();
  for (int s = 128; s > 0; s >>= 1) {
    if (threadIdx.x < s) red[threadIdx.x] += red[threadIdx.x + s];
    __syncthreads();
  }
  if (threadIdx.x == 0) out[row] = red[0];
}

// ---------------------------------------------------------------------------
extern "C" void kernel_launch(void* const* d_in, const int* in_sizes, int n_in,
                              void* d_out, int out_size, void* d_ws, size_t ws_size,
                              hipStream_t stream) {
  const float* x  = (const float*)d_in[0];
  const float* Wq = (const float*)d_in[1];
  const float* Wk = (const float*)d_in[2];
  const float* Wv = (const float*)d_in[3];
  const float* Wo = (const float*)d_in[4];
  const float* kc = (const float*)d_in[5];
  const float* vc = (const float*)d_in[6];
  const int*   pos = (const int*)d_in[7];
  float* ws  = (float*)d_ws;
  float* out = (float*)d_out;

  // zero the padded-Q region (pad heads 6..15 must be 0 for WMMA B operand)
  hipMemsetAsync(ws + WS_QPAD, 0, 4096 * sizeof(float), stream);

  qkv_kernel<<<2048, 256, 0, stream>>>(x, Wq, Wk, Wv, ws);

  dim3 gridB(NCHUNK, NUM_KV);
  decode_kernel<<<gridB, 32, 0, stream>>>(kc, vc, pos, ws);

  reduce_kernel<<<NUM_HEADS, 256, 0, stream>>>(ws);
  out_proj_kernel<<<HDIM, 256, 0, stream>>>(Wo, ws, out);
}